// GraphConvModule_63007170232986
// MI455X (gfx1250) — compile-verified
//
#include <hip/hip_runtime.h>
#include <stdint.h>

// GraphConv: out[n,c] = mean over edges e with seg_ids[e]==n of x[idxn[e],c]*w[e,c]
// Memory-bound (~475 MB HBM traffic -> ~20us at 23.3 TB/s). Strategy:
//  - wave32: 1 wave owns 128 contiguous edges, each lane owns 2 of 64 channels (float2)
//  - sorted seg_ids -> register accumulation per run, atomic flush only on seg change
//  - weights streamed via async global->LDS double buffer (ASYNCcnt pipeline),
//    tagged TH_LOAD_NT so the 410MB stream does not evict x (25.6MB) from the 192MB L2
//  - seg_ids/idxn streamed with nontemporal loads; x gather kept RT + prefetch
#define CCH 64
#define EPW 128              // edges per wave
#define WPB 8                // waves per block
#define BLOCK (WPB * 32)

__global__ void gc_zero_kernel(float* __restrict__ sums, int n_sums,
                               float* __restrict__ counts, int n_counts) {
    int i = blockIdx.x * blockDim.x + threadIdx.x;
    int stride = gridDim.x * blockDim.x;
    for (int k = i; k < n_sums; k += stride)   sums[k] = 0.0f;
    for (int k = i; k < n_counts; k += stride) counts[k] = 0.0f;
}

__global__ __launch_bounds__(BLOCK) void gc_edge_kernel(
    const float* __restrict__ x, const float* __restrict__ w,
    const int* __restrict__ idxn, const int* __restrict__ seg_ids,
    float* __restrict__ sums, float* __restrict__ counts, int n_edges) {

    __shared__ __align__(16) float wbuf[WPB][2][CCH];   // 4KB: per-wave double buffer

    const int lane = threadIdx.x & 31;
    const int wiw  = threadIdx.x >> 5;
    const long wave = (long)blockIdx.x * WPB + wiw;
    long e0 = wave * (long)EPW;
    if (e0 >= n_edges) return;
    long e1 = e0 + EPW;
    if (e1 > n_edges) e1 = n_edges;

    // Low 32 bits of a generic pointer to LDS == workgroup-relative LDS byte offset
    const uint32_t lds_s0 = (uint32_t)(uintptr_t)(&wbuf[wiw][0][0]) + (uint32_t)lane * 8u;
    const uint32_t lds_s1 = lds_s0 + (uint32_t)CCH * 4u;

    auto issue_async = [&](long e, int slot) {
        uint64_t g = (uint64_t)(uintptr_t)(w + (size_t)e * CCH + lane * 2);
        uint32_t l = slot ? lds_s1 : lds_s0;
        // each lane copies 8B -> one wave copies the full 256B weight row to LDS.
        // NT: weights are a use-once 410MB stream; keep them out of far caches so
        // the gathered x rows stay resident in the 192MB L2.
        asm volatile("global_load_async_to_lds_b64 %0, %1, off th:TH_LOAD_NT"
                     :: "v"(l), "v"(g) : "memory");
    };

    auto flush = [&](int seg, float2 a, float c) {
        float* p = sums + (size_t)seg * CCH + lane * 2;
        atomicAdd(p,     a.x);
        atomicAdd(p + 1, a.y);
        if (lane == 0) atomicAdd(counts + seg, c);
    };

    // Prologue: fill both pipeline slots
    issue_async(e0, 0);
    if (e0 + 1 < e1) issue_async(e0 + 1, 1);

    float2 acc; acc.x = 0.0f; acc.y = 0.0f;
    float cnt = 0.0f;
    int cur = __builtin_nontemporal_load(seg_ids + e0);

    for (long e = e0; e < e1; ++e) {
        int slot = (int)(e - e0) & 1;
        int seg = __builtin_nontemporal_load(seg_ids + e);   // uniform across the wave
        if (seg != cur) {
            flush(cur, acc, cnt);
            acc.x = 0.0f; acc.y = 0.0f; cnt = 0.0f;
            cur = seg;
        }
        int idx = __builtin_nontemporal_load(idxn + e);

        // Async loads complete in-order: if a newer one is outstanding, wait<=1
        // still guarantees the copy for edge e has landed in LDS.
        if (e + 1 < e1) asm volatile("s_wait_asynccnt 0x1" ::: "memory");
        else            asm volatile("s_wait_asynccnt 0x0" ::: "memory");

        const float2 wv = *(const float2*)(&wbuf[wiw][slot][lane * 2]);
        const float2 xv = *(const float2*)(x + (size_t)idx * CCH + lane * 2);
        acc.x = fmaf(xv.x, wv.x, acc.x);
        acc.y = fmaf(xv.y, wv.y, acc.y);
        cnt += 1.0f;

        if (e + 2 < e1) {
            issue_async(e + 2, slot);         // recycle this slot (ds read already consumed)
            int pidx = __builtin_nontemporal_load(idxn + e + 2);
            __builtin_prefetch(x + (size_t)pidx * CCH + lane * 2, 0, 1);
        }
    }
    flush(cur, acc, cnt);
}

__global__ void gc_finalize_kernel(float* __restrict__ out,
                                   const float* __restrict__ counts, int total) {
    int i = blockIdx.x * blockDim.x + threadIdx.x;
    if (i >= total) return;
    float d = counts[i >> 6];                 // i / CCH
    float v = out[i];
    out[i] = (d > 0.0f) ? (v / d) : 0.0f;     // deg integral: max(deg,1)==deg when deg>0
}

extern "C" void kernel_launch(void* const* d_in, const int* in_sizes, int n_in,
                              void* d_out, int out_size, void* d_ws, size_t ws_size,
                              hipStream_t stream) {
    const float* x    = (const float*)d_in[0];
    const float* w    = (const float*)d_in[1];
    const int* idxn   = (const int*)d_in[2];
    const int* segids = (const int*)d_in[3];

    float* sums   = (float*)d_out;           // accumulate sums directly in d_out
    float* counts = (float*)d_ws;            // n_nodes floats of scratch

    const int n_edges = in_sizes[2];
    const int n_nodes = in_sizes[0] / CCH;
    const int total   = n_nodes * CCH;

    gc_zero_kernel<<<1024, 256, 0, stream>>>(sums, total, counts, n_nodes);

    long waves = ((long)n_edges + EPW - 1) / EPW;
    int blocks = (int)((waves + WPB - 1) / WPB);
    gc_edge_kernel<<<blocks, BLOCK, 0, stream>>>(x, w, idxn, segids,
                                                 sums, counts, n_edges);

    gc_finalize_kernel<<<(total + 255) / 256, 256, 0, stream>>>(sums, counts, total);
}